// RandLANeighborFea_65592740544736
// MI455X (gfx1250) — compile-verified
//
#include <hip/hip_runtime.h>

// KNN (K=16) relative-feature kernel for MI455X (gfx1250, wave32).
// Per 16x16 tile: V_WMMA_F32_16X16X4_F32 with
//   A = [-2x, -2y, -2z, 1]  (16x4, row i)
//   B = [ x,   y,   z, |xj|^2 ]^T (4x16, col j)
//   C = |xi|^2 (loop-invariant!)
// so D = |xi|^2 + |xj|^2 - 2<xi,xj> = squared distances, no per-tile VALU.

typedef __attribute__((ext_vector_type(2))) float v2f;
typedef __attribute__((ext_vector_type(8))) float v8f;

#define BATCH  4
#define NPTS   8192
#define KNN    16
#define JC     512           // J-chunk staged in LDS
#define WAVES  8
#define TILE_M 128           // WAVES * 16 query rows per block

__global__ __launch_bounds__(256) void
RandLANeighborFea_kernel(const float* __restrict__ xyz,
                         float* __restrict__ feat,     // [B,N,K,10]
                         float* __restrict__ idxOut)   // [B,N,K] as float
{
    // Operand-ordered chunk: [ x(0..JC) | z(JC..2JC) | y(2JC..3JC) | sq(3JC..4JC) ]
    // -> B vgpr0 = chunk[half*JC + col], B vgpr1 = chunk[2*JC + half*JC + col]
    __shared__ float chunkArr[4 * JC];                         //  8 KB
    __shared__ __align__(16) float tileBuf[WAVES * 512];       // 16 KB (2 tiles/wave)
    __shared__ float candD[256 * KNN];                         // 16 KB
    __shared__ int   candI[256 * KNN];                         // 16 KB

    const int tid  = threadIdx.x;
    const int wave = tid >> 5;
    const int lane = tid & 31;
    const int half = lane >> 4;      // 0: lanes 0-15 (K0/K1), 1: lanes 16-31 (K2/K3)
    const int row  = lane & 15;      // M (for A) / col N (for B,D)

    const int batch = blockIdx.x / (NPTS / TILE_M);
    const int mtile = blockIdx.x % (NPTS / TILE_M);
    const int i0    = mtile * TILE_M + wave * 16;   // first query row of wave

    const float* xb = xyz + (size_t)batch * NPTS * 3;

    // ---- A operand (loop-invariant): lane holds row M=lane&15.
    //      vgpr0 = K0(-2x) / K2(-2z), vgpr1 = K1(-2y) / K3(1.0)
    v2f amat;
    {
        const int qa = i0 + row;
        const float qx = xb[qa * 3 + 0];
        const float qy = xb[qa * 3 + 1];
        const float qz = xb[qa * 3 + 2];
        amat[0] = -2.0f * (half ? qz : qx);
        amat[1] = half ? 1.0f : (-2.0f * qy);
    }

    // ---- C operand (loop-invariant): |x_i|^2 for D-rows (row = v + 8*half)
    v8f cvec;
#pragma unroll
    for (int v = 0; v < 8; ++v) {
        const int qi = i0 + v + 8 * half;
        const float qx = xb[qi * 3 + 0];
        const float qy = xb[qi * 3 + 1];
        const float qz = xb[qi * 3 + 2];
        cvec[v] = qx * qx + qy * qy + qz * qz;
    }

    // ---- per-lane top-K (sorted ascending) over this lane's half-rows
    float bestD[KNN];
    int   bestI[KNN];
#pragma unroll
    for (int q = 0; q < KNN; ++q) { bestD[q] = 3.0e38f; bestI[q] = 0; }

    for (int ch = 0; ch < NPTS / JC; ++ch) {
        const int j0 = ch * JC;
        __syncthreads();                       // previous chunk fully consumed
#pragma unroll
        for (int p = 0; p < JC / 256; ++p) {   // cooperative chunk load (SoA, operand order)
            const int jj = p * 256 + tid;
            const float* s = xb + (size_t)(j0 + jj) * 3;
            const float px = s[0], py = s[1], pz = s[2];
            chunkArr[0 * JC + jj] = px;
            chunkArr[1 * JC + jj] = pz;
            chunkArr[2 * JC + jj] = py;
            chunkArr[3 * JC + jj] = px * px + py * py + pz * pz;
        }
        __syncthreads();

#pragma unroll 1
        for (int jt = 0; jt < JC / 16; jt += 2) {
            const int jb0 = jt * 16 + row;     // this lane's B column, tile 0
            const int jb1 = jb0 + 16;          // tile 1
            v2f bm0, bm1;
            bm0[0] = chunkArr[half * JC + jb0];
            bm0[1] = chunkArr[2 * JC + half * JC + jb0];
            bm1[0] = chunkArr[half * JC + jb1];
            bm1[1] = chunkArr[2 * JC + half * JC + jb1];

            v8f d2a = __builtin_amdgcn_wmma_f32_16x16x4_f32(
                false, amat, false, bm0, (short)0, cvec, false, false);
            v8f d2b = __builtin_amdgcn_wmma_f32_16x16x4_f32(
                false, amat, false, bm1, (short)0, cvec, false, false);

            // stage both tiles: element (v + 8*half, row)
            float* t0 = tileBuf + wave * 512;
            float* t1 = t0 + 256;
#pragma unroll
            for (int v = 0; v < 8; ++v) {
                t0[(v + 8 * half) * 16 + row] = d2a[v];
                t1[(v + 8 * half) * 16 + row] = d2b[v];
            }
            // LDS ops from one wave are processed in order (ISA 7.3): a later
            // ds_load sees earlier ds_stores from any lane; only block compiler motion.
            asm volatile("" ::: "memory");

            // each lane reads its own row-half: row=lane&15, cols half*8..+8
            const float4* rp0 = (const float4*)(t0 + row * 16 + half * 8);
            const float4* rp1 = (const float4*)(t1 + row * 16 + half * 8);
            const float4 a0 = rp0[0], a1 = rp0[1];
            const float4 b0 = rp1[0], b1 = rp1[1];
            const float dv[16] = { a0.x, a0.y, a0.z, a0.w,  a1.x, a1.y, a1.z, a1.w,
                                   b0.x, b0.y, b0.z, b0.w,  b1.x, b1.y, b1.z, b1.w };
            const int jgb = j0 + jt * 16 + half * 8;
#pragma unroll
            for (int cc = 0; cc < 16; ++cc) {
                const float d  = dv[cc];
                const int   jg = jgb + cc + ((cc >= 8) ? 8 : 0);  // tile1 is +16 cols
                if (d < bestD[KNN - 1]) {      // rare: sorted insertion
                    float dk = d; int ik = jg;
#pragma unroll
                    for (int q = 0; q < KNN; ++q) {
                        const bool sw = dk < bestD[q];
                        const float td = sw ? bestD[q] : dk;
                        const int   ti = sw ? bestI[q] : ik;
                        bestD[q] = sw ? dk : bestD[q];
                        bestI[q] = sw ? ik : bestI[q];
                        dk = td; ik = ti;
                    }
                }
            }
        }
    }

    // ---- publish per-lane lists, then merge the two half-row lists per row
    {
        const int cb = tid * KNN;
#pragma unroll
        for (int q = 0; q < KNN; ++q) { candD[cb + q] = bestD[q]; candI[cb + q] = bestI[q]; }
    }
    asm volatile("s_wait_dscnt 0" ::: "memory");

    if (half == 0) {
        const int i = i0 + row;                 // query index within batch
        const float qx = xb[i * 3 + 0];
        const float qy = xb[i * 3 + 1];
        const float qz = xb[i * 3 + 2];
        const int baseA = tid * KNN;            // cols 0-7 (mod 16) stream
        const int baseB = (tid + 16) * KNN;     // cols 8-15 (mod 16) stream
        int pa = 0, pb = 0;
        const size_t rowOut = ((size_t)batch * NPTS + i) * KNN;
        for (int k = 0; k < KNN; ++k) {
            const float da = candD[baseA + pa];
            const float db = candD[baseB + pb];
            const bool ta = da <= db;
            const float d = ta ? da : db;
            const int  nb = ta ? candI[baseA + pa] : candI[baseB + pb];
            pa += ta ? 1 : 0;
            pb += ta ? 0 : 1;

            const float* ns = xb + (size_t)nb * 3;
            const float nx = ns[0], ny = ns[1], nz = ns[2];

            float* o = feat + (rowOut + k) * 10;
            const float2 f0 = { d,       qx - nx };
            const float2 f1 = { qy - ny, qz - nz };
            const float2 f2 = { qx,      qy      };
            const float2 f3 = { qz,      nx      };
            const float2 f4 = { ny,      nz      };
            *(float2*)(o + 0) = f0;
            *(float2*)(o + 2) = f1;
            *(float2*)(o + 4) = f2;
            *(float2*)(o + 6) = f3;
            *(float2*)(o + 8) = f4;
            idxOut[rowOut + k] = (float)nb;
        }
    }
}

extern "C" void kernel_launch(void* const* d_in, const int* in_sizes, int n_in,
                              void* d_out, int out_size, void* d_ws, size_t ws_size,
                              hipStream_t stream) {
    (void)in_sizes; (void)n_in; (void)out_size; (void)d_ws; (void)ws_size;
    const float* xyz = (const float*)d_in[0];
    float* out    = (float*)d_out;
    float* idxOut = out + (size_t)BATCH * NPTS * KNN * 10;   // second tuple output
    dim3 grid(BATCH * (NPTS / TILE_M));
    RandLANeighborFea_kernel<<<grid, 256, 0, stream>>>(xyz, out, idxOut);
}